// ProjectionOnly_83940840833165
// MI455X (gfx1250) — compile-verified
//
#include <hip/hip_runtime.h>

// Problem sizes (fixed by the reference)
#define B_    64
#define D_    2048
#define J_    196      // 14*14
#define C_    128
#define NCLS_ 1000

typedef __attribute__((ext_vector_type(2))) float v2f;
typedef __attribute__((ext_vector_type(8))) float v8f;

// ---------------------------------------------------------------------------
// Kernel 0: a_sq[c] = sum_d W_land[c,d]^2   (tiny)
// ---------------------------------------------------------------------------
__global__ void k_asq(const float* __restrict__ Wl, float* __restrict__ a_sq) {
  const int c = threadIdx.x;
  if (c < C_) {
    const float* wp = Wl + (size_t)c * D_;
    float s = 0.f;
    for (int d = 0; d < D_; ++d) { const float v = wp[d]; s += v * v; }
    a_sq[c] = s;
  }
}

// ---------------------------------------------------------------------------
// Kernel 1: per (b, j-tile of 16): ab = x^T @ W_land^T via V_WMMA_F32_16X16X4_F32,
// then maps = 2*ab - ||x_j||^2 - ||w_c||^2, softmax over c, write m[b,c,j].
// Block = 256 threads (8 waves); wave w owns c-tile [16w, 16w+16).
// ---------------------------------------------------------------------------
__global__ __launch_bounds__(256) void k_maps(
    const float* __restrict__ x, const float* __restrict__ Wl,
    const float* __restrict__ a_sq, float* __restrict__ m) {
  __shared__ float s_ab[16][C_];   // 16 j rows x 128 c  (8 KB)
  __shared__ float s_bsq[32];
  __shared__ float s_asq[C_];

  const int b    = blockIdx.y;
  const int j0   = blockIdx.x * 16;
  const int tid  = threadIdx.x;
  const int lane = tid & 31;
  const int wave = tid >> 5;
  const int nl   = lane & 15;   // M for A-frag (=j), N for B-frag (=c)
  const int hi   = lane >> 4;   // lane half selects K pair
  const int kb   = hi * 2;
  const int c0   = wave * 16;

  if (tid < C_) s_asq[tid] = a_sq[tid];

  const int j  = j0 + nl;
  const int jc = (j < J_) ? j : (J_ - 1);   // clamp: padded j never goes OOB

  const float* xb = x  + (size_t)b * D_ * J_ + jc;          // + d*J
  const float* wp = Wl + (size_t)(c0 + nl) * D_ + kb;       // + d (contiguous pair)

  v8f acc = {};
  float bsq = 0.f;
  for (int d0 = 0; d0 < D_; d0 += 4) {
    // A: x^T[j, d0+kb .. d0+kb+1]  (stride J between the two -> 2x b32)
    const float a0 = xb[(size_t)(d0 + kb) * J_];
    const float a1 = xb[(size_t)(d0 + kb + 1) * J_];
    v2f A = {a0, a1};
    // B: W_land[c, d0+kb .. +1]  contiguous -> global_load_b64
    v2f Bv = *(const v2f*)(wp + d0);
    acc = __builtin_amdgcn_wmma_f32_16x16x4_f32(false, A, false, Bv,
                                                (short)0, acc, false, false);
    bsq += a0 * a0 + a1 * a1;   // partial ||x_j||^2 (kept by wave 0)
  }
  if (wave == 0) s_bsq[lane] = bsq;

  // Scatter C/D fragment into LDS panel: element (M = v + 8*hi, N = c0+nl)
  const int cc = c0 + nl;
#pragma unroll
  for (int v = 0; v < 8; ++v) s_ab[v + hi * 8][cc] = acc[v];
  __syncthreads();

  // Softmax over c for each valid j row (16 rows, one thread each)
  if (tid < 16 && (j0 + tid) < J_) {
    const float bs = s_bsq[tid] + s_bsq[tid + 16];
    float mx = -3.4e38f;
#pragma unroll 4
    for (int c = 0; c < C_; ++c) {
      const float v = 2.0f * s_ab[tid][c] - bs - s_asq[c];
      s_ab[tid][c] = v;
      mx = fmaxf(mx, v);
    }
    float sum = 0.f;
#pragma unroll 4
    for (int c = 0; c < C_; ++c) {
      const float e = __expf(s_ab[tid][c] - mx);
      s_ab[tid][c] = e;
      sum += e;
    }
    const float inv = 1.0f / sum;
    float* mp = m + (size_t)b * C_ * J_ + (j0 + tid);
#pragma unroll 4
    for (int c = 0; c < C_; ++c) mp[(size_t)c * J_] = s_ab[tid][c] * inv;
  }
}

// ---------------------------------------------------------------------------
// Kernel 2: updates[b,c,d] = sum_j m[b,c,j] * x[b,d,j]
// Grid (4 d-supers, 8 c-tiles, B). Block 256 = 8 waves; wave owns 16c x 64d
// (4 accumulators, A fragment reused 4x per k-step).
// ---------------------------------------------------------------------------
__global__ __launch_bounds__(256) void k_updates(
    const float* __restrict__ x, const float* __restrict__ m,
    float* __restrict__ upd) {
  const int b    = blockIdx.z;
  const int c0   = blockIdx.y * 16;
  const int tid  = threadIdx.x;
  const int lane = tid & 31;
  const int wave = tid >> 5;
  const int nl   = lane & 15;
  const int hi   = lane >> 4;
  const int kb   = hi * 2;
  const int dbase = blockIdx.x * 512 + wave * 64;

  const float* mp = m + (size_t)b * C_ * J_ + (size_t)(c0 + nl) * J_ + kb;
  const float* xb = x + (size_t)b * D_ * J_ + kb;

  v8f acc0 = {}, acc1 = {}, acc2 = {}, acc3 = {};
  for (int j0 = 0; j0 < J_; j0 += 4) {   // 196 = 49*4 exactly
    v2f A  = *(const v2f*)(mp + j0);                                    // b64
    v2f B0 = *(const v2f*)(xb + (size_t)(dbase +  0 + nl) * J_ + j0);   // b64
    v2f B1 = *(const v2f*)(xb + (size_t)(dbase + 16 + nl) * J_ + j0);
    v2f B2 = *(const v2f*)(xb + (size_t)(dbase + 32 + nl) * J_ + j0);
    v2f B3 = *(const v2f*)(xb + (size_t)(dbase + 48 + nl) * J_ + j0);
    acc0 = __builtin_amdgcn_wmma_f32_16x16x4_f32(false, A, false, B0, (short)0, acc0, false, false);
    acc1 = __builtin_amdgcn_wmma_f32_16x16x4_f32(false, A, false, B1, (short)0, acc1, false, false);
    acc2 = __builtin_amdgcn_wmma_f32_16x16x4_f32(false, A, false, B2, (short)0, acc2, false, false);
    acc3 = __builtin_amdgcn_wmma_f32_16x16x4_f32(false, A, false, B3, (short)0, acc3, false, false);
  }

  // Store: c = c0 + v + 8*hi, d = dbase + 16*t + nl (coalesced over lanes)
  float* up = upd + ((size_t)b * C_ + c0 + hi * 8) * D_ + dbase + nl;
#pragma unroll
  for (int v = 0; v < 8; ++v) {
    up[(size_t)v * D_ +  0] = acc0[v];
    up[(size_t)v * D_ + 16] = acc1[v];
    up[(size_t)v * D_ + 32] = acc2[v];
    up[(size_t)v * D_ + 48] = acc3[v];
  }
}

// ---------------------------------------------------------------------------
// Kernel 3: cpt[b,c] = mean_j m[b,c,j]; also duplicate into cpt_r
// ---------------------------------------------------------------------------
__global__ void k_cpt(const float* __restrict__ m, float* __restrict__ cpt,
                      float* __restrict__ cpt_r) {
  const int i = blockIdx.x * blockDim.x + threadIdx.x;   // b*C + c
  if (i < B_ * C_) {
    const float* mp = m + (size_t)i * J_;
    float s = 0.f;
#pragma unroll 4
    for (int jj = 0; jj < J_; ++jj) s += mp[jj];
    const float v = s * (1.0f / (float)J_);
    cpt[i] = v;
    cpt_r[i] = v;
  }
}

// ---------------------------------------------------------------------------
// Kernel 4: scores = cpt @ W_cls^T + b_cls  (64x1000x128, tiny)
// ---------------------------------------------------------------------------
__global__ void k_scores(const float* __restrict__ cpt, const float* __restrict__ Wc,
                         const float* __restrict__ bc, float* __restrict__ scores) {
  const int i = blockIdx.x * blockDim.x + threadIdx.x;
  if (i < B_ * NCLS_) {
    const int b = i / NCLS_, n = i % NCLS_;
    const float* cp = cpt + b * C_;
    const float* wp = Wc + (size_t)n * C_;
    float s = bc[n];
#pragma unroll 8
    for (int c = 0; c < C_; ++c) s += cp[c] * wp[c];
    scores[i] = s;
  }
}

// ---------------------------------------------------------------------------
extern "C" void kernel_launch(void* const* d_in, const int* in_sizes, int n_in,
                              void* d_out, int out_size, void* d_ws, size_t ws_size,
                              hipStream_t stream) {
  const float* x  = (const float*)d_in[0];   // [B,D,H,W]
  const float* Wl = (const float*)d_in[1];   // [C,D]
  const float* Wc = (const float*)d_in[2];   // [NCLS,C]
  const float* bc = (const float*)d_in[3];   // [NCLS]

  float* out    = (float*)d_out;
  float* cpt    = out;                        // [B,C]            8192
  float* scores = out + 8192;                 // [B,NCLS]        64000
  float* cpt_r  = out + 8192 + 64000;         // [B,C,1]          8192
  float* upd    = out + 8192 + 64000 + 8192;  // [B,C,D]      16777216

  float* a_sq = (float*)d_ws;                 // 128 floats
  float* m    = (float*)d_ws + 256;           // [B,C,J] fp32, ~6.42 MB, 1KB-aligned

  k_asq<<<1, 128, 0, stream>>>(Wl, a_sq);
  k_maps<<<dim3(13, B_), 256, 0, stream>>>(x, Wl, a_sq, m);   // 13*16 = 208 >= 196
  k_updates<<<dim3(4, 8, B_), 256, 0, stream>>>(x, m, upd);
  k_cpt<<<(B_ * C_ + 255) / 256, 256, 0, stream>>>(m, cpt, cpt_r);
  k_scores<<<(B_ * NCLS_ + 255) / 256, 256, 0, stream>>>(cpt, Wc, bc, scores);
}